// Pool_11501922418833
// MI455X (gfx1250) — compile-verified
//
#include <hip/hip_runtime.h>
#include <hip/hip_bf16.h>

#define G_   32
#define N_   2048
#define D_   128
#define K_   1024
#define NT_  (G_ * N_)    // 65536 nodes
#define NP_  (G_ * K_)    // 32768 kept nodes
#define E_   (G_ * 16384) // 524288 edges

typedef float v2f __attribute__((ext_vector_type(2)));
typedef float v8f __attribute__((ext_vector_type(8)));

// ---------------------------------------------------------------------------
// Kernel 1: per-node dot products against {w_rel, w_root, w_lin} via fp32 WMMA.
// GEMM view: [NT_ x 128] * [128 x 16] with B cols 0..2 = the three weight
// vectors, cols 3..15 zero. One wave handles 16 nodes: 32x V_WMMA_F32_16X16X4_F32.
// ---------------------------------------------------------------------------
__global__ __launch_bounds__(128) void k_dots(const float* __restrict__ x,
                                              const float* __restrict__ w_rel,
                                              const float* __restrict__ w_root,
                                              const float* __restrict__ w_lin,
                                              float* __restrict__ xw /* [3][NT_] */) {
    __shared__ float sB[D_ * 16];   // B[k][n], row-major, 8 KB
    const int tid = threadIdx.x;
    {   // build B: one thread per k-row
        const int k = tid;          // blockDim.x == 128 == D_
        sB[k * 16 + 0] = w_rel[k];
        sB[k * 16 + 1] = w_root[k];
        sB[k * 16 + 2] = w_lin[k];
#pragma unroll
        for (int n = 3; n < 16; ++n) sB[k * 16 + n] = 0.0f;
    }
    __syncthreads();

    const int lane = tid & 31;
    const int wave = tid >> 5;
    const int base = blockIdx.x * 64 + wave * 16;   // first node of this wave's tile
    const int m    = lane & 15;                     // row within tile (A) / col (B,D)
    const int half = lane >> 4;                     // K-half selector per ISA layout

    // A layout (32-bit 16x4): lanes 0-15 hold {K=0,K=1}, lanes 16-31 hold {K=2,K=3}
    const float* xrow = x + (size_t)(base + m) * D_ + half * 2;

    v8f c = {};
#pragma unroll 4
    for (int k0 = 0; k0 < D_; k0 += 4) {
        v2f a = *(const v2f*)(xrow + k0);           // 8B aligned
        const int kk = k0 + half * 2;
        v2f b;
        b.x = sB[kk * 16 + m];
        b.y = sB[(kk + 1) * 16 + m];
        // (neg_a, A, neg_b, B, c_mod, C, reuse_a, reuse_b)
        c = __builtin_amdgcn_wmma_f32_16x16x4_f32(false, a, false, b,
                                                  (short)0, c, false, false);
    }

    // D layout: VGPR j -> row M = j + 8*half, col N = m. Cols 0..2 are our dots.
    if (m < 3) {
        float* outcol = xw + (size_t)m * NT_ + base + half * 8;
#pragma unroll
        for (int j = 0; j < 8; ++j) outcol[j] = c[j];
    }
}

// ---------------------------------------------------------------------------
__global__ void k_zero(float* __restrict__ p, int n) {
    int i = blockIdx.x * blockDim.x + threadIdx.x;
    if (i < n) p[i] = 0.0f;
}

// scalar edge scatter: agg[dst] += (x[src] . w_rel)
__global__ void k_scatter(const int* __restrict__ src, const int* __restrict__ dst,
                          const float* __restrict__ xw_rel, float* __restrict__ agg) {
    int e = blockIdx.x * blockDim.x + threadIdx.x;
    if (e < E_) atomicAdd(&agg[dst[e]], xw_rel[src[e]]);
}

__global__ void k_score(const float* __restrict__ xw, const float* __restrict__ agg,
                        const float* __restrict__ b_rel, const float* __restrict__ b_lin,
                        float* __restrict__ score) {
    int i = blockIdx.x * blockDim.x + threadIdx.x;
    if (i < NT_) {
        float s1 = agg[i] + b_rel[0] + xw[NT_ + i];      // GraphConv score
        float s2 = xw[2 * NT_ + i] + b_lin[0];           // Linear score
        score[i] = fmaxf(s1, s2);
    }
}

// ---------------------------------------------------------------------------
// Per-graph bitonic sort of 2048 (score, idx) pairs in LDS; comparator =
// (score desc, index asc) which exactly reproduces jax.lax.top_k ordering.
// ---------------------------------------------------------------------------
__global__ __launch_bounds__(1024) void k_topk(const float* __restrict__ score,
                                               int* __restrict__ perm,
                                               float* __restrict__ score_perm,
                                               int* __restrict__ new_idx,
                                               float* __restrict__ out_batch) {
    __shared__ float sk[N_];
    __shared__ int   si[N_];
    const int g = blockIdx.x;
    const int t = threadIdx.x;
    const float* sc = score + g * N_;

    for (int i = t; i < N_; i += 1024) { sk[i] = sc[i]; si[i] = i; }

    for (int size = 2; size <= N_; size <<= 1) {
        for (int stride = size >> 1; stride > 0; stride >>= 1) {
            __syncthreads();
            int i = ((t & ~(stride - 1)) << 1) | (t & (stride - 1));
            int j = i + stride;
            bool dir = ((i & size) == 0);
            float ki = sk[i], kj = sk[j];
            int   ii = si[i], ij = si[j];
            bool jFirst = (kj > ki) || (kj == ki && ij < ii);
            if (jFirst == dir) { sk[i] = kj; sk[j] = ki; si[i] = ij; si[j] = ii; }
        }
    }
    __syncthreads();

    for (int pos = t; pos < N_; pos += 1024) {
        int nl = si[pos];
        int ng = g * N_ + nl;
        if (pos < K_) {
            int o = g * K_ + pos;
            perm[o]       = ng;
            score_perm[o] = sk[pos];
            new_idx[ng]   = o;
            out_batch[o]  = (float)g;
        } else {
            new_idx[ng] = -1;
        }
    }
}

// x_out[row] = x[perm[row]] * tanh(score_perm[row]); 32 lanes * float4 per row
__global__ __launch_bounds__(256) void k_gather(const float* __restrict__ x,
                                                const int* __restrict__ perm,
                                                const float* __restrict__ score_perm,
                                                float* __restrict__ x_out) {
    int tid  = blockIdx.x * 256 + threadIdx.x;
    int row  = tid >> 5;
    int lane = tid & 31;
    if (row >= NP_) return;
    int p = perm[row];
    float tg = tanhf(score_perm[row]);
    const float4* xin = (const float4*)(x + (size_t)p * D_);
    float4*       xo  = (float4*)(x_out + (size_t)row * D_);
    float4 v = xin[lane];
    v.x *= tg; v.y *= tg; v.z *= tg; v.w *= tg;
    xo[lane] = v;
}

__global__ void k_edges(const int* __restrict__ src, const int* __restrict__ dst,
                        const int* __restrict__ new_idx,
                        float* __restrict__ out_e, float* __restrict__ out_mask) {
    int e = blockIdx.x * blockDim.x + threadIdx.x;
    if (e >= E_) return;
    int ns = new_idx[src[e]];
    int nd = new_idx[dst[e]];
    bool ok = (ns >= 0) && (nd >= 0);
    out_e[e]      = ok ? (float)ns : 0.0f;
    out_e[E_ + e] = ok ? (float)nd : 0.0f;
    out_mask[e]   = ok ? 1.0f : 0.0f;
}

// ---------------------------------------------------------------------------
extern "C" void kernel_launch(void* const* d_in, const int* in_sizes, int n_in,
                              void* d_out, int out_size, void* d_ws, size_t ws_size,
                              hipStream_t stream) {
    const float* x       = (const float*)d_in[0];
    const int*   eidx    = (const int*)d_in[1];
    // d_in[2] = batch (unused; new_batch derivable as graph id)
    const float* w_rel   = (const float*)d_in[3];
    const float* b_rel   = (const float*)d_in[4];
    const float* w_root  = (const float*)d_in[5];
    const float* w_lin   = (const float*)d_in[6];
    const float* b_lin   = (const float*)d_in[7];
    const int* src = eidx;
    const int* dst = eidx + E_;

    char* ws = (char*)d_ws;
    float* xw         = (float*)ws; ws += sizeof(float) * 3 * NT_;
    float* agg        = (float*)ws; ws += sizeof(float) * NT_;
    float* score      = (float*)ws; ws += sizeof(float) * NT_;
    int*   perm       = (int*)ws;   ws += sizeof(int)   * NP_;
    float* score_perm = (float*)ws; ws += sizeof(float) * NP_;
    int*   new_idx    = (int*)ws;   ws += sizeof(int)   * NT_;

    float* out       = (float*)d_out;
    float* out_x     = out;                         // NP_ * D_
    float* out_e     = out_x + (size_t)NP_ * D_;    // 2 * E_
    float* out_mask  = out_e + 2 * (size_t)E_;      // E_
    float* out_batch = out_mask + (size_t)E_;       // NP_

    k_dots   <<<NT_ / 64,        128,  0, stream>>>(x, w_rel, w_root, w_lin, xw);
    k_zero   <<<(NT_ + 255)/256, 256,  0, stream>>>(agg, NT_);
    k_scatter<<<(E_ + 255)/256,  256,  0, stream>>>(src, dst, xw, agg);
    k_score  <<<(NT_ + 255)/256, 256,  0, stream>>>(xw, agg, b_rel, b_lin, score);
    k_topk   <<<G_,              1024, 0, stream>>>(score, perm, score_perm, new_idx, out_batch);
    k_gather <<<(NP_*32 + 255)/256, 256, 0, stream>>>(x, perm, score_perm, out_x);
    k_edges  <<<(E_ + 255)/256,  256,  0, stream>>>(src, dst, new_idx, out_e, out_mask);
}